// MultiHeadAttention_19980187861824
// MI455X (gfx1250) — compile-verified
//
#include <hip/hip_runtime.h>
#include <hip/hip_bf16.h>
#include <math.h>

// Problem constants from the reference.
#define BB 2
#define SS 2048
#define DD 1024
#define HH 16
#define ADIM 64

typedef __attribute__((ext_vector_type(2))) float v2f;
typedef __attribute__((ext_vector_type(8))) float v8f;

// ---------------------------------------------------------------------------
// Generic fp32 WMMA GEMM: C[z] = A[z] (MxK, row-major) * B[z] (KxN, row-major)
// + bias (optional). One wave (32 lanes) computes one 16x16 output tile using
// V_WMMA_F32_16X16X4_F32. M, N multiples of 16; K multiple of 4.
// A-matrix lane layout (ISA 7.12.2, 32-bit A 16x4):
//   lanes 0-15: row=lane,   VGPR0=K0, VGPR1=K1
//   lanes16-31: row=lane-16, VGPR0=K2, VGPR1=K3   -> contiguous 8B load
// B-matrix (4x16): VGPR0 holds rows {K0 (lanes0-15), K2 (lanes16-31)},
//                  VGPR1 holds rows {K1, K3}; column = lane&15.
// C/D: VGPR j -> row j + 8*(lane>=16), col = lane&15.
// ---------------------------------------------------------------------------
__launch_bounds__(32)
__global__ void wmma_gemm_f32_kernel(const float* __restrict__ A,
                                     const float* __restrict__ B,
                                     const float* __restrict__ bias,
                                     float* __restrict__ C,
                                     int M, int N, int K,
                                     long long sA, long long sB, long long sC) {
  const int lane = threadIdx.x & 31;
  const int half = lane >> 4;     // 0 or 1
  const int l16  = lane & 15;
  const int col0 = blockIdx.x * 16;
  const int row0 = blockIdx.y * 16;

  const float* Ab = A + (long long)blockIdx.z * sA;
  const float* Bb = B + (long long)blockIdx.z * sB;
  float*       Cb = C + (long long)blockIdx.z * sC;

  const long long arow = (long long)(row0 + l16) * (long long)K;

  v8f acc = {};
  for (int kk = 0; kk < K; kk += 4) {
    const int ka = kk + 2 * half;
    v2f a = *(const v2f*)(Ab + arow + ka);          // 8B aligned (ka even)
    v2f b;
    b.x = Bb[(long long)ka * N + col0 + l16];
    b.y = Bb[(long long)(ka + 1) * N + col0 + l16];
    acc = __builtin_amdgcn_wmma_f32_16x16x4_f32(
        /*neg_a=*/false, a, /*neg_b=*/false, b,
        /*c_mod=*/(short)0, acc, /*reuse_a=*/false, /*reuse_b=*/false);
  }

  const float bv = bias ? bias[col0 + l16] : 0.0f;
#pragma unroll
  for (int j = 0; j < 8; ++j) {
    const int r = row0 + j + 8 * half;
    Cb[(long long)r * N + col0 + l16] = acc[j] + bv;
  }
}

// ---------------------------------------------------------------------------
// Fused attention: for one (batch, 16-query-row tile), loop heads:
//   energy stripe 16xS via WMMA fp32 (QKt), scale 1/8, mask, softmax in LDS,
//   accumulate probs/H into the global head-averaged attention matrix.
// 8 waves: waves split the 128 column tiles for energy; each wave owns 2 rows
// for the softmax. 16xS fp32 score stripe = 128 KB LDS (CDNA5: 320 KB/WGP).
// ---------------------------------------------------------------------------
__launch_bounds__(256)
__global__ void attn_softmax_kernel(const float* __restrict__ Q,
                                    const float* __restrict__ Kmat,
                                    const int* __restrict__ mask,
                                    float* __restrict__ attn) {
  __shared__ float sc[16 * SS];   // 131072 bytes

  const int b    = blockIdx.x / (SS / 16);
  const int qt   = blockIdx.x % (SS / 16);
  const int q0   = qt * 16;
  const int lane = threadIdx.x & 31;
  const int wave = threadIdx.x >> 5;   // 0..7
  const int half = lane >> 4;
  const int l16  = lane & 15;

  const float* Qb = Q    + (size_t)b * SS * DD;
  const float* Kb = Kmat + (size_t)b * SS * DD;
  float*    attnB = attn + (size_t)b * SS * SS;

  for (int h = 0; h < HH; ++h) {
    const float* Qh = Qb + h * ADIM;
    const float* Kh = Kb + h * ADIM;

    // ---- energy stripe: 128 column tiles of 16, split across 8 waves ----
    for (int ct = wave; ct < SS / 16; ct += 8) {
      const int k0 = ct * 16;
      const float* ap = Qh + (size_t)(q0 + l16) * DD + 2 * half;
      const float* bp = Kh + (size_t)(k0 + l16) * DD + 2 * half;
      v8f e = {};
#pragma unroll
      for (int kk = 0; kk < ADIM; kk += 4) {
        v2f a  = *(const v2f*)(ap + kk);   // Q[q0+l16][h*64 + kk + 2*half ..]
        v2f bb = *(const v2f*)(bp + kk);   // K[k0+l16][h*64 + kk + 2*half ..] (B = K^T)
        e = __builtin_amdgcn_wmma_f32_16x16x4_f32(
            false, a, false, bb, (short)0, e, false, false);
      }
#pragma unroll
      for (int j = 0; j < 8; ++j) {
        const int qr = j + 8 * half;                 // row within stripe
        float v = e[j] * 0.125f;                     // 1/sqrt(64)
        if (mask[(size_t)(q0 + qr) * SS + k0 + l16] == 0) v = -1.0e10f;
        sc[qr * SS + k0 + l16] = v;
      }
    }
    __syncthreads();

    // ---- softmax over each of the 16 rows; wave w owns rows 2w, 2w+1 ----
#pragma unroll
    for (int rr = 0; rr < 2; ++rr) {
      const int r = wave * 2 + rr;
      float m = -3.0e38f;
      for (int k = lane; k < SS; k += 32) m = fmaxf(m, sc[r * SS + k]);
      for (int off = 16; off; off >>= 1) m = fmaxf(m, __shfl_xor(m, off, 32));
      float s = 0.0f;
      for (int k = lane; k < SS; k += 32) {
        const float ex = __expf(sc[r * SS + k] - m);
        sc[r * SS + k] = ex;
        s += ex;
      }
      for (int off = 16; off; off >>= 1) s += __shfl_xor(s, off, 32);
      const float inv = 1.0f / (s * (float)HH);      // softmax + mean over heads
      const size_t rowBase = (size_t)(q0 + r) * SS;
      if (h == 0) {
        for (int k = lane; k < SS; k += 32)
          attnB[rowBase + k] = sc[r * SS + k] * inv;
      } else {
        for (int k = lane; k < SS; k += 32)
          attnB[rowBase + k] += sc[r * SS + k] * inv;
      }
    }
    __syncthreads();
  }
}

// ---------------------------------------------------------------------------
// kernel_launch: Q/K/V projections -> fused attention -> attn*V -> *Wh + bh
// d_out layout: out [B,S,D] (4,194,304 f32) then attention [B,S,S] (8,388,608).
// Workspace: Q(16MB) K(16MB) V(1MB) ctx(1MB) ~= 34 MB.
// ---------------------------------------------------------------------------
extern "C" void kernel_launch(void* const* d_in, const int* in_sizes, int n_in,
                              void* d_out, int out_size, void* d_ws, size_t ws_size,
                              hipStream_t stream) {
  (void)in_sizes; (void)n_in; (void)out_size; (void)ws_size;

  const float* query = (const float*)d_in[0];
  const float* key   = (const float*)d_in[1];
  const float* value = (const float*)d_in[2];
  const int*   mask  = (const int*)d_in[3];
  const float* Wq    = (const float*)d_in[4];
  const float* bq    = (const float*)d_in[5];
  const float* Wk    = (const float*)d_in[6];
  const float* bk    = (const float*)d_in[7];
  const float* Wv    = (const float*)d_in[8];
  const float* bv    = (const float*)d_in[9];
  const float* Wh    = (const float*)d_in[10];
  const float* bh    = (const float*)d_in[11];

  float* out  = (float*)d_out;                         // [B*S, D]
  float* attn = out + (size_t)BB * SS * DD;            // [B, S, S]

  float* ws = (float*)d_ws;
  float* Qw = ws;                                      // [B*S, D]
  float* Kw = Qw + (size_t)BB * SS * DD;               // [B*S, D]
  float* Vw = Kw + (size_t)BB * SS * DD;               // [B, S, AD]
  float* Cx = Vw + (size_t)BB * SS * ADIM;             // [B, S, AD]

  const dim3 wblk(32);

  // Q = query @ Wq + bq   (M=4096, N=1024, K=1024)
  wmma_gemm_f32_kernel<<<dim3(DD / 16, (BB * SS) / 16, 1), wblk, 0, stream>>>(
      query, Wq, bq, Qw, BB * SS, DD, DD, 0, 0, 0);
  // K = key @ Wk + bk
  wmma_gemm_f32_kernel<<<dim3(DD / 16, (BB * SS) / 16, 1), wblk, 0, stream>>>(
      key, Wk, bk, Kw, BB * SS, DD, DD, 0, 0, 0);
  // V = value @ Wv + bv   (N=64)
  wmma_gemm_f32_kernel<<<dim3(ADIM / 16, (BB * SS) / 16, 1), wblk, 0, stream>>>(
      value, Wv, bv, Vw, BB * SS, ADIM, DD, 0, 0, 0);

  // attention = mean_h softmax(mask(QK^T/8))   (writes d_out attention region)
  attn_softmax_kernel<<<dim3(BB * (SS / 16)), dim3(256), 0, stream>>>(
      Qw, Kw, mask, attn);

  // ctx = attention @ V    (batched: M=2048, N=64, K=2048)
  wmma_gemm_f32_kernel<<<dim3(ADIM / 16, SS / 16, BB), wblk, 0, stream>>>(
      attn, Vw, nullptr, Cx, SS, ADIM, SS,
      (long long)SS * SS, (long long)SS * ADIM, (long long)SS * ADIM);

  // out = ctx @ Wh + bh    (M=4096, N=1024, K=64)
  wmma_gemm_f32_kernel<<<dim3(DD / 16, (BB * SS) / 16, 1), wblk, 0, stream>>>(
      Cx, Wh, bh, out, BB * SS, DD, ADIM, 0, 0, 0);
}